// Hadamard_2156073583155
// MI455X (gfx1250) — compile-verified
//
#include <hip/hip_runtime.h>
#include <hip/hip_bf16.h>

typedef __attribute__((ext_vector_type(2))) float v2f;
typedef __attribute__((ext_vector_type(8))) float v8f;

// H16[a][b] = (-1)^popcount(a & b)  (Sylvester / natural Hadamard ordering,
// which matches the reference's bit-by-bit butterfly network exactly).
__device__ __forceinline__ float hsign(int a, int b) {
    return (__popc(a & b) & 1) ? -1.0f : 1.0f;
}

#define WMMA_F32X4(a, b, c) \
    __builtin_amdgcn_wmma_f32_16x16x4_f32(false, (a), false, (b), (short)0, (c), false, false)

// One workgroup = one row of 8192 fp32.  8 waves (wave32), 32 KB LDS.
// Factorization: H8192 = (H2 (x) H16)_[bits 12..8] (x) H16_[bits 7..4] (x) H16_[bits 3..0]
__global__ __launch_bounds__(256) void wht8192_kernel(const float* __restrict__ x,
                                                      float* __restrict__ y) {
    __shared__ float lds[8192];

    const int row  = blockIdx.x;
    const int tid  = threadIdx.x;
    const int lane = tid & 31;
    const int wave = tid >> 5;        // 0..7
    const int half = (lane >> 4) & 1; // lane-half select for WMMA layouts
    const int l16  = lane & 15;

    // ---------- global -> LDS (coalesced 16B) ----------
    {
        const float4* gx = (const float4*)(x + (size_t)row * 8192u);
        float4* l4 = (float4*)lds;
#pragma unroll
        for (int i = 0; i < 8; ++i)
            l4[i * 256 + tid] = gx[i * 256 + tid];
    }
    __syncthreads();

    // ---------- Stage 1: transform index bits [3:0].  Per 16x16 tile: X <- X * H16
    // A (16x4 f32 layout): lane l16 = M; vgpr0/1 hold K = k0, k0+1 with k0 = 4*kk + 2*half.
    // B slice = H16 rows, generated in VALU.  D is C-layout, stored back in place.
#pragma unroll
    for (int tt = 0; tt < 4; ++tt) {
        float* base = lds + (wave * 4 + tt) * 256;
        v8f c = {};
#pragma unroll
        for (int kk = 0; kk < 4; ++kk) {
            const int k0 = kk * 4 + 2 * half;
            v2f a, b;
            a.x = base[l16 * 16 + k0];
            a.y = base[l16 * 16 + k0 + 1];
            b.x = hsign(k0, l16);
            b.y = hsign(k0 + 1, l16);
            c = WMMA_F32X4(a, b, c);
        }
#pragma unroll
        for (int v = 0; v < 8; ++v)
            base[(v + 8 * half) * 16 + l16] = c[v]; // C-layout: M = v + 8*half, N = l16
    }
    // No barrier: each wave owns its 4 tiles through stages 1 and 2 (per-wave DS is in-order).

    // ---------- Stage 2: transform index bits [7:4].  Per tile: X <- H16 * X
#pragma unroll
    for (int tt = 0; tt < 4; ++tt) {
        float* base = lds + (wave * 4 + tt) * 256;
        v8f c = {};
#pragma unroll
        for (int kk = 0; kk < 4; ++kk) {
            const int k0 = kk * 4 + 2 * half;
            v2f a, b;
            a.x = hsign(l16, k0);
            a.y = hsign(l16, k0 + 1);
            b.x = base[k0 * 16 + l16];
            b.y = base[(k0 + 1) * 16 + l16];
            c = WMMA_F32X4(a, b, c);
        }
#pragma unroll
        for (int v = 0; v < 8; ++v)
            base[(v + 8 * half) * 16 + l16] = c[v];
    }
    __syncthreads();

    // ---------- Stage 3: transform index bits [12:8] (32 tiles, stride 256).
    // H32 = H2 (x) H16: the H2 butterfly across tile halves (t, t+16) is folded
    // into the B-operand load (bu = top+bot, bv = top-bot); H16 applied by WMMA.
    // Wave w owns flat columns [w*32, w*32+32) -> in-place update is race-free.
#pragma unroll
    for (int cc = 0; cc < 2; ++cc) {
        const int ch = wave * 2 + cc; // 16-column chunk within the 256-wide tile face
        v8f cu = {}, cv = {};
#pragma unroll
        for (int kk = 0; kk < 4; ++kk) {
            const int k0 = kk * 4 + 2 * half;
            v2f a, bu, bv;
            a.x = hsign(l16, k0);
            a.y = hsign(l16, k0 + 1);
            const float t0 = lds[(k0)      * 256 + ch * 16 + l16];
            const float t1 = lds[(k0 + 1)  * 256 + ch * 16 + l16];
            const float b0 = lds[(k0 + 16) * 256 + ch * 16 + l16];
            const float b1 = lds[(k0 + 17) * 256 + ch * 16 + l16];
            bu.x = t0 + b0; bu.y = t1 + b1;
            bv.x = t0 - b0; bv.y = t1 - b1;
            cu = WMMA_F32X4(a, bu, cu);
            cv = WMMA_F32X4(a, bv, cv);
        }
#pragma unroll
        for (int v = 0; v < 8; ++v) {
            const int m = v + 8 * half;
            lds[(m)      * 256 + ch * 16 + l16] = cu[v]; // output tile index m      (top half)
            lds[(m + 16) * 256 + ch * 16 + l16] = cv[v]; // output tile index m + 16 (bottom)
        }
    }
    __syncthreads();

    // ---------- LDS -> global, scaled by 1/sqrt(8192), coalesced 16B ----------
    {
        const float scale = 0.011048543456039806f; // 2^-6.5
        float4* gy = (float4*)(y + (size_t)row * 8192u);
        const float4* l4 = (const float4*)lds;
#pragma unroll
        for (int i = 0; i < 8; ++i) {
            float4 v = l4[i * 256 + tid];
            v.x *= scale; v.y *= scale; v.z *= scale; v.w *= scale;
            gy[i * 256 + tid] = v;
        }
    }
}

extern "C" void kernel_launch(void* const* d_in, const int* in_sizes, int n_in,
                              void* d_out, int out_size, void* d_ws, size_t ws_size,
                              hipStream_t stream) {
    (void)in_sizes; (void)n_in; (void)out_size; (void)d_ws; (void)ws_size;
    const float* x = (const float*)d_in[0];
    float* y = (float*)d_out;
    wht8192_kernel<<<dim3(8192), dim3(256), 0, stream>>>(x, y);
}